// SpatialAttention_Deform_LKA_sequential_13417477833356
// MI455X (gfx1250) — compile-verified
//
#include <hip/hip_runtime.h>
#include <math.h>
#include <stdint.h>

// ---------------- fixed problem dims (from setup_inputs) ----------------
constexpr int BB = 2, CC = 128, SS = 20, NN = 8000, NH = 4, DH = 32, PP = 32;
constexpr int C3 = 3 * CC;  // 384
constexpr int KTAPS = 27;
constexpr int OC_PAD = 96;  // offset conv: 81 out channels padded to 96

typedef __attribute__((ext_vector_type(2))) float v2f;
typedef __attribute__((ext_vector_type(8))) float v8f;
typedef unsigned int u32x4 __attribute__((ext_vector_type(4)));
typedef int i32x4 __attribute__((ext_vector_type(4)));
typedef int i32x8 __attribute__((ext_vector_type(8)));

enum { EPI_NONE = 0, EPI_GELU = 1, EPI_ACCUM = 2, EPI_RESID = 3 };

// ---- Tensor Data Mover: 1-D contiguous copy global -> LDS (ISA ch.8) ----
__device__ inline void tdm_load_1d(const float* gsrc, unsigned lds_off,
                                   unsigned nElem) {
  unsigned long long ga = (unsigned long long)(uintptr_t)gsrc;
  u32x4 g0;
  g0[0] = 1u;                                   // count=1 (valid descriptor)
  g0[1] = lds_off;                              // lds_addr (bytes)
  g0[2] = (unsigned)(ga & 0xFFFFFFFFu);         // global_addr[31:0]
  g0[3] = (unsigned)((ga >> 32) & 0x01FFFFFFu)  // global_addr[56:32]
          | (2u << 30);                         // type=2 ("image")
  i32x8 g1;
  g1[0] = (int)(2u << 16);                      // wg_mask=0, data_size=4B
  g1[1] = (int)((nElem & 0xFFFFu) << 16);       // tensor_dim0 lo16
  g1[2] = (int)((nElem >> 16) & 0xFFFFu) | (1 << 16);  // dim0 hi16, dim1=1
  g1[3] = (int)((nElem & 0xFFFFu) << 16);       // tile_dim0 (= whole tensor)
  g1[4] = 1;                                    // tile_dim1=1, tile_dim2=0
  g1[5] = (int)nElem;                           // tensor_dim0_stride lo32
  g1[6] = 0;
  g1[7] = 0;
  i32x4 z4 = {0, 0, 0, 0};
#if __clang_major__ >= 23
  i32x8 z8 = {0, 0, 0, 0, 0, 0, 0, 0};
  __builtin_amdgcn_tensor_load_to_lds(g0, g1, z4, z4, z8, 0);
#else
  __builtin_amdgcn_tensor_load_to_lds(g0, g1, z4, z4, 0);
#endif
}

__device__ inline v8f wmma4(v2f a, v2f b, v8f c) {
  return __builtin_amdgcn_wmma_f32_16x16x4_f32(false, a, false, b, (short)0, c,
                                               false, false);
}

template <int MODE, long RSD, long CSD>
__device__ inline void epi_store(float* Db, const float* Rb, v8f acc,
                                 long mbase, long n, int h, float alpha,
                                 float bv) {
#pragma unroll
  for (int i = 0; i < 8; ++i) {
    const long mr = mbase + i + 8 * h;       // C/D VGPR layout (ISA 7.12.2)
    const long di = mr * RSD + n * CSD;
    float v = acc[i] * alpha + bv;
    if constexpr (MODE == EPI_GELU)
      v = 0.5f * v * (1.0f + erff(v * 0.70710678118654752440f));
    if constexpr (MODE == EPI_ACCUM) v += Db[di];
    if constexpr (MODE == EPI_RESID) v += Rb[di];
    Db[di] = v;
  }
}

// =======================================================================
// Batched f32 GEMM on V_WMMA_F32_16X16X4_F32, compile-time strides.
// Wave32 computes a 32x32 tile (4 accumulators, 16 WMMAs / unrolled iter).
// A(m,k)=A[m*RSA+k*CSA], B(k,n)=B[k*RSB+n*CSB], D(m,n)=D[m*RSD+n*CSD].
// Requires M%32==0, Nout%32==0, Kdim%16==0 (call sites guarantee this).
// STAGE: B panel (Nout*Kdim floats, contiguous since RSB==1,CSB==Kdim)
// is DMA'd to LDS by the Tensor Data Mover once per block, then read
// with ds_load_b64 -- B is reused by all 8 waves of the block.
// =======================================================================
template <long RSA, long CSA, long RSB, long CSB, long RSD, long CSD, int MODE,
          bool STAGE>
__global__ void __launch_bounds__(256) gemm_wmma(
    const float* __restrict__ A, const float* __restrict__ Bm,
    float* __restrict__ D, const float* __restrict__ bias,
    const float* __restrict__ Res, const float* __restrict__ alphas,
    int alphaMod, long bsAo, long bsAi, long bsBo, long bsBi, long bsDo,
    long bsDi, int bDiv, int mt, int nt, int Kdim) {
  const int wave = threadIdx.x >> 5;
  const int tile = blockIdx.x * 8 + wave;
  const bool active = (tile < mt * nt);
  const int z = blockIdx.y;
  const int zo = z / bDiv, zi = z - zo * bDiv;
  const float* Bb = Bm + zo * bsBo + zi * bsBi;

  const int lane = threadIdx.x & 31;
  const int h = lane >> 4;  // K half (ISA 16x4 A / 4x16 B lane split)
  const int l = lane & 15;

  const int tdiv = active ? tile : 0;
  const int tm = tdiv / nt, tn = tdiv % nt;
  const long m0 = (long)tm * 32 + l, m1 = m0 + 16;
  const long n0 = (long)tn * 32 + l, n1 = n0 + 16;

  const float* Ab = A + zo * bsAo + zi * bsAi;
  const float* A0 = Ab + m0 * RSA;
  const float* A1 = Ab + m1 * RSA;

  v8f acc00 = {0, 0, 0, 0, 0, 0, 0, 0}, acc01 = acc00, acc10 = acc00,
      acc11 = acc00;

  if constexpr (STAGE) {
    extern __shared__ float ldsB[];
    if (threadIdx.x < 32) {
      tdm_load_1d(Bb, (unsigned)(uintptr_t)&ldsB[0],
                  (unsigned)((long)nt * 32 * Kdim));
      __builtin_amdgcn_s_wait_tensorcnt(0);
    }
    __syncthreads();
    if (!active) return;
    const float* B0 = &ldsB[0] + n0 * (long)Kdim;
    const float* B1 = &ldsB[0] + n1 * (long)Kdim;
#pragma unroll 4
    for (int kb = 0; kb < Kdim; kb += 4) {
      const long k0 = kb + 2 * h;
      const v2f a0 = *(const v2f*)(A0 + k0);
      const v2f a1 = *(const v2f*)(A1 + k0);
      const v2f b0 = *(const v2f*)(B0 + k0);   // ds_load_b64
      const v2f b1 = *(const v2f*)(B1 + k0);
      acc00 = wmma4(a0, b0, acc00);
      acc01 = wmma4(a0, b1, acc01);
      acc10 = wmma4(a1, b0, acc10);
      acc11 = wmma4(a1, b1, acc11);
    }
  } else {
    if (!active) return;
    const float* B0 = Bb + n0 * CSB;
    const float* B1 = Bb + n1 * CSB;
#pragma unroll 4
    for (int kb = 0; kb < Kdim; kb += 4) {
      const long k0 = kb + 2 * h;
      v2f a0, a1, b0, b1;
      if constexpr (CSA == 1) {
        a0 = *(const v2f*)(A0 + k0);
        a1 = *(const v2f*)(A1 + k0);
      } else {
        a0.x = A0[k0 * CSA];
        a0.y = A0[k0 * CSA + CSA];
        a1.x = A1[k0 * CSA];
        a1.y = A1[k0 * CSA + CSA];
      }
      if constexpr (RSB == 1) {
        b0 = *(const v2f*)(B0 + k0);
        b1 = *(const v2f*)(B1 + k0);
      } else {
        b0.x = B0[k0 * RSB];
        b0.y = B0[k0 * RSB + RSB];
        b1.x = B1[k0 * RSB];
        b1.y = B1[k0 * RSB + RSB];
      }
      acc00 = wmma4(a0, b0, acc00);
      acc01 = wmma4(a0, b1, acc01);
      acc10 = wmma4(a1, b0, acc10);
      acc11 = wmma4(a1, b1, acc11);
    }
  }

  const float alpha = (alphas != nullptr) ? alphas[z % alphaMod] : 1.0f;
  const float bv0 = (bias != nullptr) ? bias[n0] : 0.0f;
  const float bv1 = (bias != nullptr) ? bias[n1] : 0.0f;
  float* Db = D + zo * bsDo + zi * bsDi;
  const float* Rb = (Res != nullptr) ? (Res + zo * bsDo + zi * bsDi) : nullptr;
  epi_store<MODE, RSD, CSD>(Db, Rb, acc00, (long)tm * 32, n0, h, alpha, bv0);
  epi_store<MODE, RSD, CSD>(Db, Rb, acc01, (long)tm * 32, n1, h, alpha, bv1);
  epi_store<MODE, RSD, CSD>(Db, Rb, acc10, (long)tm * 32 + 16, n0, h, alpha, bv0);
  epi_store<MODE, RSD, CSD>(Db, Rb, acc11, (long)tm * 32 + 16, n1, h, alpha, bv1);
}

template <long RSA, long CSA, long RSB, long CSB, long RSD, long CSD, int MODE,
          bool STAGE>
static void launch_gemm(hipStream_t s, const float* A, const float* Bm,
                        float* D, const float* bias, const float* Res,
                        const float* alphas, int alphaMod, long bsAo, long bsAi,
                        long bsBo, long bsBi, long bsDo, long bsDi, int bDiv,
                        int nbatch, int M, int Nout, int Kdim) {
  const int mt = M / 32, nt = Nout / 32;
  dim3 grid((mt * nt + 7) / 8, nbatch);
  const size_t shmem = STAGE ? (size_t)Nout * Kdim * 4 : 0;
  gemm_wmma<RSA, CSA, RSB, CSB, RSD, CSD, MODE, STAGE><<<grid, 256, shmem, s>>>(
      A, Bm, D, bias, Res, alphas, alphaMod, bsAo, bsAi, bsBo, bsBi, bsDo,
      bsDi, bDiv, mt, nt, Kdim);
}

// ---------------- weight prep (one-shot layout transforms) ----------------
__global__ void prep_dw_k(const float* __restrict__ w, float* __restrict__ wT) {
  const int i = blockIdx.x * 256 + threadIdx.x;  // 125*128
  if (i >= 125 * CC) return;
  const int c = i & 127, t = i >> 7;
  wT[t * CC + c] = w[c * 125 + t];
}
__global__ void prep_off_k(const float* __restrict__ w, float* __restrict__ wt) {
  const long i = (long)blockIdx.x * 256 + threadIdx.x;  // 27*96*128
  if (i >= (long)KTAPS * OC_PAD * CC) return;
  const int c = (int)(i & 127);
  const int oc = (int)((i >> 7) % OC_PAD);
  const int t = (int)(i / ((long)OC_PAD * CC));
  wt[i] = (oc < 81) ? w[(long)oc * (CC * KTAPS) + c * KTAPS + t] : 0.0f;
}
__global__ void prep_def_k(const float* __restrict__ w, float* __restrict__ wt) {
  const long i = (long)blockIdx.x * 256 + threadIdx.x;  // 27*128*128
  if (i >= (long)KTAPS * CC * CC) return;
  const int c = (int)(i & 127);
  const int o = (int)((i >> 7) & 127);
  const int t = (int)(i / ((long)CC * CC));
  wt[i] = w[(long)o * (CC * KTAPS) + c * KTAPS + t];
}

// ---- split qkv rows -> qA (bh,N,dh), kT/vT (bh,dh,N) -------------------
__global__ void split_qkv_k(const float* __restrict__ qkv,
                            float* __restrict__ qA, float* __restrict__ kT,
                            float* __restrict__ vT) {
  const long i = (long)blockIdx.x * 256 + threadIdx.x;  // B*N*C
  if (i >= (long)BB * NN * CC) return;
  const int c = (int)(i & 127);
  const long bn = i >> 7;
  const int n = (int)(bn % NN);
  const int b = (int)(bn / NN);
  const int hh = c >> 5, d = c & 31;
  const int bh = b * NH + hh;
  const long base = bn * C3 + c;
  qA[((long)bh * NN + n) * DH + d] = qkv[base];
  kT[((long)bh * DH + d) * NN + n] = qkv[base + CC];
  vT[((long)bh * DH + d) * NN + n] = qkv[base + 2 * CC];
}

// ------- q /= max(||q||_2 over N, 1e-12); qA layout (bh, N, dh) ----------
__global__ void qnorm_k(float* __restrict__ qA) {
  const int r = blockIdx.x;  // bh*32 + d, 256 rows
  const int bh = r >> 5, d = r & 31;
  float* p = qA + (long)bh * NN * DH + d;
  float ss = 0.f;
  for (int n = threadIdx.x; n < NN; n += 256) {
    const float v = p[(long)n * DH];
    ss += v * v;
  }
  __shared__ float sbuf[256];
  sbuf[threadIdx.x] = ss;
  __syncthreads();
  for (int s = 128; s > 0; s >>= 1) {
    if ((int)threadIdx.x < s) sbuf[threadIdx.x] += sbuf[threadIdx.x + s];
    __syncthreads();
  }
  const float inv = 1.0f / fmaxf(sqrtf(sbuf[0]), 1e-12f);
  for (int n = threadIdx.x; n < NN; n += 256) p[(long)n * DH] *= inv;
}

// ------------------------- softmax over len-32 rows -----------------------
__global__ void softmax32_k(float* __restrict__ attn, long rows) {
  const long r = (long)blockIdx.x * 256 + threadIdx.x;
  if (r >= rows) return;
  float* p = attn + r * PP;
  float mx = -3.4e38f;
#pragma unroll
  for (int j = 0; j < PP; ++j) mx = fmaxf(mx, p[j]);
  float s = 0.f;
#pragma unroll
  for (int j = 0; j < PP; ++j) {
    const float e = expf(p[j] - mx);
    p[j] = e;
    s += e;
  }
  const float inv = 1.0f / s;
#pragma unroll
  for (int j = 0; j < PP; ++j) p[j] *= inv;
}

// --------------- LayerNorm over contiguous rows of 128 (wave/row) ---------
__global__ void ln128_k(float* __restrict__ x, const float* __restrict__ g,
                        const float* __restrict__ bt, long rows) {
  const long r = (long)blockIdx.x * 8 + (threadIdx.x >> 5);
  if (r >= rows) return;
  const int lane = threadIdx.x & 31;
  float* p = x + r * CC;
  float v[4];
  float s = 0.f;
#pragma unroll
  for (int j = 0; j < 4; ++j) {
    v[j] = p[lane + 32 * j];
    s += v[j];
  }
  for (int o = 16; o > 0; o >>= 1) s += __shfl_xor(s, o, 32);
  const float mean = s * (1.0f / 128.0f);
  float var = 0.f;
#pragma unroll
  for (int j = 0; j < 4; ++j) {
    const float d = v[j] - mean;
    var += d * d;
  }
  for (int o = 16; o > 0; o >>= 1) var += __shfl_xor(var, o, 32);
  const float inv = rsqrtf(var * (1.0f / 128.0f) + 1e-5f);
#pragma unroll
  for (int j = 0; j < 4; ++j) {
    const int c = lane + 32 * j;
    p[c] = (v[j] - mean) * inv * g[c] + bt[c];
  }
}

// ------------- depthwise 5x5x5 conv, NHWC, dilation + padding -------------
__global__ void dwconv5_k(const float* __restrict__ in,
                          const float* __restrict__ wT,  // (125, C)
                          const float* __restrict__ bias,
                          float* __restrict__ out, int dil, int pad) {
  const long i = (long)blockIdx.x * 256 + threadIdx.x;  // B*N*C
  if (i >= (long)BB * NN * CC) return;
  const int c = (int)(i & 127);
  const long bn = i >> 7;
  const int v = (int)(bn % NN);
  const int b = (int)(bn / NN);
  const int x3 = v % SS, x2 = (v / SS) % SS, x1 = v / (SS * SS);
  const float* ib = in + (long)b * NN * CC + c;
  float acc = bias[c];
  for (int t1 = 0; t1 < 5; ++t1) {
    const int p1 = x1 + t1 * dil - pad;
    if ((unsigned)p1 >= (unsigned)SS) continue;
    for (int t2 = 0; t2 < 5; ++t2) {
      const int p2 = x2 + t2 * dil - pad;
      if ((unsigned)p2 >= (unsigned)SS) continue;
      for (int t3 = 0; t3 < 5; ++t3) {
        const int p3 = x3 + t3 * dil - pad;
        if ((unsigned)p3 >= (unsigned)SS) continue;
        acc += wT[((t1 * 5 + t2) * 5 + t3) * CC + c] *
               ib[(long)((p1 * SS + p2) * SS + p3) * CC];
      }
    }
  }
  out[i] = acc;
}

// ------- zero-padded spatial shift (one 3x3x3 tap), NHWC ------------------
__global__ void shift3_k(const float* __restrict__ in, float* __restrict__ out,
                         int dz, int dy, int dx) {
  const long i = (long)blockIdx.x * 256 + threadIdx.x;  // B*N*C
  if (i >= (long)BB * NN * CC) return;
  const int c = (int)(i & 127);
  const long bn = i >> 7;
  const int v = (int)(bn % NN);
  const int b = (int)(bn / NN);
  const int x3 = v % SS, x2 = (v / SS) % SS, x1 = v / (SS * SS);
  const int p1 = x1 + dz, p2 = x2 + dy, p3 = x3 + dx;
  float val = 0.f;
  if ((unsigned)p1 < (unsigned)SS && (unsigned)p2 < (unsigned)SS &&
      (unsigned)p3 < (unsigned)SS)
    val = in[((long)b * NN + (p1 * SS + p2) * SS + p3) * CC + c];
  out[i] = val;
}

// ------- trilinear deform sampling for one tap (NHWC) ---------------------
__global__ void deform_sample_k(const float* __restrict__ x,
                                const float* __restrict__ off,  // (B,N,96)
                                float* __restrict__ sampk, int ktap) {
  const long bv = blockIdx.x;  // B*N blocks, 128 threads = channels
  const int v = (int)(bv % NN);
  const int b = (int)(bv / NN);
  const int x3 = v % SS, x2 = (v / SS) % SS, x1 = v / (SS * SS);
  const int k1 = ktap / 9 - 1, k2 = (ktap / 3) % 3 - 1, k3 = ktap % 3 - 1;
  const float* ob = off + bv * OC_PAD + ktap * 3;
  const float p1 = (float)(x1 + k1) + ob[0];
  const float p2 = (float)(x2 + k2) + ob[1];
  const float p3 = (float)(x3 + k3) + ob[2];
  const float f1 = floorf(p1), f2 = floorf(p2), f3 = floorf(p3);
  const float r1 = p1 - f1, r2 = p2 - f2, r3 = p3 - f3;
  const int i1 = (int)f1, i2 = (int)f2, i3 = (int)f3;
  int idx[8];
  float wt[8];
#pragma unroll
  for (int d = 0; d < 8; ++d) {
    const int d1 = d >> 2, d2 = (d >> 1) & 1, d3 = d & 1;
    const int q1 = i1 + d1, q2 = i2 + d2, q3 = i3 + d3;
    const bool valid =
        (q1 >= 0 && q1 < SS && q2 >= 0 && q2 < SS && q3 >= 0 && q3 < SS);
    const float w0 = (d1 ? r1 : 1.f - r1) * (d2 ? r2 : 1.f - r2) *
                     (d3 ? r3 : 1.f - r3);
    wt[d] = valid ? w0 : 0.f;
    const int c1 = min(max(q1, 0), SS - 1);
    const int c2 = min(max(q2, 0), SS - 1);
    const int c3 = min(max(q3, 0), SS - 1);
    idx[d] = (c1 * SS + c2) * SS + c3;
  }
  const int c = threadIdx.x;
  const float* xb = x + (long)b * NN * CC + c;
  float acc = 0.f;
#pragma unroll
  for (int d = 0; d < 8; ++d) acc += wt[d] * xb[(long)idx[d] * CC];
  sampk[bv * CC + c] = acc;
}

// ------------------------------ y = u * a --------------------------------
__global__ void mul_k(const float* __restrict__ u, const float* __restrict__ a,
                      float* __restrict__ y, long n) {
  const long i = (long)blockIdx.x * 256 + threadIdx.x;
  if (i < n) y[i] = u[i] * a[i];
}

// ---- double reshape/transpose scramble (NHWC in, NHWC out) --------------
__global__ void rescramble_k(const float* __restrict__ y1,
                             float* __restrict__ y2) {
  const long i = (long)blockIdx.x * 256 + threadIdx.x;  // B*N*C
  if (i >= (long)BB * NN * CC) return;
  const int c2 = (int)(i & 127);
  const long rem = i >> 7;
  const int n2 = (int)(rem % NN);
  const int b = (int)(rem / NN);
  const long t = (long)c2 * NN + n2;
  const int n = (int)(t >> 7);
  const int c = (int)(t & 127);
  y2[i] = y1[((long)b * NN + n) * CC + c];
}

// ========================= host-side orchestration ========================
extern "C" void kernel_launch(void* const* d_in, const int* in_sizes, int n_in,
                              void* d_out, int out_size, void* d_ws,
                              size_t ws_size, hipStream_t stream) {
  (void)in_sizes; (void)n_in; (void)out_size; (void)ws_size;
  const float* x        = (const float*)d_in[0];
  const float* qkv_w    = (const float*)d_in[1];
  const float* E_w      = (const float*)d_in[2];
  const float* E_b      = (const float*)d_in[3];
  const float* temp     = (const float*)d_in[4];
  const float* norm_w   = (const float*)d_in[5];
  const float* norm_b   = (const float*)d_in[6];
  const float* proj1_w  = (const float*)d_in[7];
  const float* proj1_b  = (const float*)d_in[8];
  const float* conv0_w  = (const float*)d_in[9];
  const float* conv0_b  = (const float*)d_in[10];
  const float* convsp_w = (const float*)d_in[11];
  const float* convsp_b = (const float*)d_in[12];
  const float* offset_w = (const float*)d_in[13];
  const float* offset_b = (const float*)d_in[14];
  const float* deform_w = (const float*)d_in[15];
  const float* deform_b = (const float*)d_in[16];
  const float* conv1_w  = (const float*)d_in[17];
  const float* conv1_b  = (const float*)d_in[18];
  const float* proj2_w  = (const float*)d_in[19];
  const float* proj2_b  = (const float*)d_in[20];
  const float* norm2_w  = (const float*)d_in[21];
  const float* norm2_b  = (const float*)d_in[22];
  const float* out_w    = (const float*)d_in[23];
  const float* out_b    = (const float*)d_in[24];
  float* out = (float*)d_out;
  float* w = (float*)d_ws;

  // workspace layout (floats), with buffer reuse
  const long F_QKV  = 0;                    // 16000*384 = 6,144,000
  const long F_QA   = 6144000;              // (bh,N,dh) 2,048,000
  const long F_KT   = 8192000;              // (bh,dh,N)
  const long F_VT   = 10240000;
  const long F_KP   = 12288000;             // (bh,p,d) 8192
  const long F_VP   = 12296192;             // (bh,d,p) 8192
  const long F_ATTN = 12304384;             // (bh,n,p) 2,048,000
  const long F_XSA  = 14352384;             // scrambled (B,N,C); = shortcut
  const long F_YC   = 16400384;             // proj2 out (B,N,C)
  const long F_Y2   = 18448384;             // rescrambled (B,N,C)
  const long F_WDW0 = 20496384;             // (125,128)
  const long F_WDW1 = F_WDW0 + 16000;
  const long F_WTO  = F_WDW1 + 16000;       // (27,96,128)
  const long F_WTD  = F_WTO + (long)KTAPS * OC_PAD * CC;  // (27,128,128)
  // reused regions (qkv / qA / kT / vT free after attention):
  const long F_YB   = F_QKV;                // gelu(proj1) = "u"
  const long F_A0   = F_QKV + 2048000;      // conv0 out (later conv1 out)
  const long F_A1   = F_QKV + 4096000;      // convsp out
  const long F_OFF  = F_KT;                 // offset conv out (B,N,96)
  const long F_XSH  = F_VT;                 // shifted-tap buffer
  const long F_SMP  = F_QA;                 // deform sampled tap
  const long F_DFO  = F_ATTN;               // deform conv out
  const long F_YA   = F_VT;                 // u*a (xsh free by then)

  const long BNC = (long)BB * NN * CC;      // 2,048,000
  const int EW_GRID = (int)((BNC + 255) / 256);
  const int M_ALL = BB * NN;                // 16000 (flattened NHWC rows)

  // weight layout prep
  prep_dw_k<<<dim3((125 * CC + 255) / 256), 256, 0, stream>>>(conv0_w, w + F_WDW0);
  prep_dw_k<<<dim3((125 * CC + 255) / 256), 256, 0, stream>>>(convsp_w, w + F_WDW1);
  prep_off_k<<<dim3((KTAPS * OC_PAD * CC + 255) / 256), 256, 0, stream>>>(
      offset_w, w + F_WTO);
  prep_def_k<<<dim3((KTAPS * CC * CC + 255) / 256), 256, 0, stream>>>(
      deform_w, w + F_WTD);

  // 1) qkv = x @ qkv_w.T   (M=16000, N=384, K=128)
  launch_gemm<CC, 1, 1, CC, C3, 1, EPI_NONE, false>(
      stream, x, qkv_w, w + F_QKV, nullptr, nullptr, nullptr, 1,
      0, 0, 0, 0, 0, 0, 1, 1, M_ALL, C3, CC);

  // 2) split; 3) q L2-normalize over tokens
  split_qkv_k<<<dim3(EW_GRID), 256, 0, stream>>>(w + F_QKV, w + F_QA, w + F_KT,
                                                 w + F_VT);
  qnorm_k<<<dim3(BB * NH * DH), 256, 0, stream>>>(w + F_QA);

  // 4) kp = kT @ E_w.T + E_b stored (bh,p,d); vp stored (bh,d,p)
  launch_gemm<NN, 1, 1, NN, 1, DH, EPI_NONE, false>(
      stream, w + F_KT, E_w, w + F_KP, E_b, nullptr, nullptr, 1,
      (long)DH * NN, 0, 0, 0, (long)DH * PP, 0, 1, BB * NH, DH, PP, NN);
  launch_gemm<NN, 1, 1, NN, DH, 1, EPI_NONE, false>(
      stream, w + F_VT, E_w, w + F_VP, E_b, nullptr, nullptr, 1,
      (long)DH * NN, 0, 0, 0, (long)DH * PP, 0, 1, BB * NH, DH, PP, NN);

  // 5) attn = (qn @ kp^T) * temperature[h]   (per bh: M=8000,N=32,K=32)
  launch_gemm<DH, 1, 1, DH, PP, 1, EPI_NONE, false>(
      stream, w + F_QA, w + F_KP, w + F_ATTN, nullptr, nullptr, temp, NH,
      (long)NN * DH, 0, (long)DH * PP, 0, (long)NN * PP, 0, 1, BB * NH,
      NN, PP, DH);

  // 6) softmax over P=32
  softmax32_k<<<dim3((BB * NH * NN + 255) / 256), 256, 0, stream>>>(
      w + F_ATTN, (long)BB * NH * NN);

  // 7) context, stored directly in the torch permute-scrambled (B,N,C) view
  launch_gemm<PP, 1, 1, PP, 1, (long)NH * NN, EPI_NONE, false>(
      stream, w + F_ATTN, w + F_VP, w + F_XSA, nullptr, nullptr, nullptr, 1,
      (long)NH * NN * PP, (long)NN * PP, (long)NH * DH * PP, (long)DH * PP,
      (long)NN * CC, NN, NH, BB * NH, NN, DH, PP);

  // 8) LayerNorm rows of C; result doubles as NHWC "xv" (residual shortcut)
  ln128_k<<<dim3((M_ALL + 7) / 8), 256, 0, stream>>>(w + F_XSA, norm_w, norm_b,
                                                     M_ALL);

  // 9) u = gelu(proj1(xv))  (TDM-staged B)
  launch_gemm<CC, 1, 1, CC, CC, 1, EPI_GELU, true>(
      stream, w + F_XSA, proj1_w, w + F_YB, proj1_b, nullptr, nullptr, 1,
      0, 0, 0, 0, 0, 0, 1, 1, M_ALL, CC, CC);

  // 10) depthwise 5^3 (pad 2) then dilated depthwise 5^3 (dil 3, pad 6)
  dwconv5_k<<<dim3(EW_GRID), 256, 0, stream>>>(w + F_YB, w + F_WDW0, conv0_b,
                                               w + F_A0, 1, 2);
  dwconv5_k<<<dim3(EW_GRID), 256, 0, stream>>>(w + F_A0, w + F_WDW1, convsp_b,
                                               w + F_A1, 3, 6);

  // 11) offset conv (C->81 pad to 96) as 27 shifted-tap GEMMs
  for (int t = 0; t < KTAPS; ++t) {
    const int dz = t / 9 - 1, dy = (t / 3) % 3 - 1, dx = t % 3 - 1;
    shift3_k<<<dim3(EW_GRID), 256, 0, stream>>>(w + F_A1, w + F_XSH, dz, dy, dx);
    const float* Bt = w + F_WTO + (long)t * OC_PAD * CC;
    if (t == 0)
      launch_gemm<CC, 1, 1, CC, OC_PAD, 1, EPI_NONE, true>(
          stream, w + F_XSH, Bt, w + F_OFF, offset_b, nullptr, nullptr, 1,
          0, 0, 0, 0, 0, 0, 1, 1, M_ALL, OC_PAD, CC);
    else
      launch_gemm<CC, 1, 1, CC, OC_PAD, 1, EPI_ACCUM, true>(
          stream, w + F_XSH, Bt, w + F_OFF, nullptr, nullptr, nullptr, 1,
          0, 0, 0, 0, 0, 0, 1, 1, M_ALL, OC_PAD, CC);
  }

  // 12) deformable conv: 27 x (trilinear sample tap -> accumulated GEMM)
  for (int t = 0; t < KTAPS; ++t) {
    deform_sample_k<<<dim3(BB * NN), dim3(CC), 0, stream>>>(
        w + F_A1, w + F_OFF, w + F_SMP, t);
    const float* Bt = w + F_WTD + (long)t * CC * CC;
    if (t == 0)
      launch_gemm<CC, 1, 1, CC, CC, 1, EPI_NONE, true>(
          stream, w + F_SMP, Bt, w + F_DFO, deform_b, nullptr, nullptr, 1,
          0, 0, 0, 0, 0, 0, 1, 1, M_ALL, CC, CC);
    else
      launch_gemm<CC, 1, 1, CC, CC, 1, EPI_ACCUM, true>(
          stream, w + F_SMP, Bt, w + F_DFO, nullptr, nullptr, nullptr, 1,
          0, 0, 0, 0, 0, 0, 1, 1, M_ALL, CC, CC);
  }

  // 13) conv1 (1x1x1)
  launch_gemm<CC, 1, 1, CC, CC, 1, EPI_NONE, true>(
      stream, w + F_DFO, conv1_w, w + F_A0, conv1_b, nullptr, nullptr, 1,
      0, 0, 0, 0, 0, 0, 1, 1, M_ALL, CC, CC);

  // 14) gate: y = u * a
  mul_k<<<dim3(EW_GRID), 256, 0, stream>>>(w + F_YB, w + F_A0, w + F_YA, BNC);

  // 15) proj2 + residual shortcut
  launch_gemm<CC, 1, 1, CC, CC, 1, EPI_RESID, true>(
      stream, w + F_YA, proj2_w, w + F_YC, proj2_b, w + F_XSA, nullptr, 1,
      0, 0, 0, 0, 0, 0, 1, 1, M_ALL, CC, CC);

  // 16) double-reshape scramble -> LayerNorm
  rescramble_k<<<dim3(EW_GRID), 256, 0, stream>>>(w + F_YC, w + F_Y2);
  ln128_k<<<dim3((M_ALL + 7) / 8), 256, 0, stream>>>(w + F_Y2, norm2_w,
                                                     norm2_b, M_ALL);

  // 17) out = y @ out_w.T + out_b  (TDM-staged B)
  launch_gemm<CC, 1, 1, CC, CC, 1, EPI_NONE, true>(
      stream, w + F_Y2, out_w, out, out_b, nullptr, nullptr, 1,
      0, 0, 0, 0, 0, 0, 1, 1, M_ALL, CC, CC);
}